// Net_67997922230811
// MI455X (gfx1250) — compile-verified
//
#include <hip/hip_runtime.h>
#include <math.h>

#define EMB 300
#define NNODES 50000
#define NEDGES 200000
#define NGRAPHS 2048
#define ATOMF 9
#define ATOMV 128
#define BONDF 3
#define BONDV 16
#define NEG_SLOPE 0.2f

typedef __attribute__((ext_vector_type(2))) float v2f;
typedef __attribute__((ext_vector_type(8))) float v8f;

static __device__ __forceinline__ float warp_sum(float v) {
#pragma unroll
  for (int off = 16; off > 0; off >>= 1) v += __shfl_xor(v, off, 32);
  return v;
}

// order-preserving uint encoding of float for atomicMax
static __device__ __forceinline__ unsigned f2key(float f) {
  unsigned u = __float_as_uint(f);
  return (u & 0x80000000u) ? ~u : (u | 0x80000000u);
}
static __device__ __forceinline__ float key2f(unsigned k) {
  unsigned u = (k & 0x80000000u) ? (k & 0x7FFFFFFFu) : ~k;
  return __uint_as_float(u);
}

// ---------------- encode: h[n][d] = sum_f atom_emb[f, x[n,f], d] -------------
__global__ void encode_atoms_kernel(const int* __restrict__ x,
                                    const float* __restrict__ table,
                                    float* __restrict__ h) {
  int n = blockIdx.x;
  __shared__ int idx[ATOMF];
  if (threadIdx.x < ATOMF) idx[threadIdx.x] = x[n * ATOMF + threadIdx.x];
  __syncthreads();
  for (int d = threadIdx.x; d < EMB; d += blockDim.x) {
    float s = 0.f;
#pragma unroll
    for (int f = 0; f < ATOMF; ++f)
      s += table[((size_t)(f * ATOMV + idx[f])) * EMB + d];
    h[(size_t)n * EMB + d] = s;
  }
}

// ---------------- fp32 WMMA GEMM: C[MxN] = A[MxK] @ B[KxN] -------------------
// K % 4 == 0, M % 16 == 0. One 16-col B panel per block (LDS, zero-padded),
// 8 waves per block, one 16x16 output tile per wave, v_wmma_f32_16x16x4_f32.
#define GEMM_WAVES 8
__global__ void gemm_wmma_kernel(const float* __restrict__ A,
                                 const float* __restrict__ B,
                                 float* __restrict__ C,
                                 int M, int K, int N) {
  __shared__ float ldsB[304 * 16];  // K <= 304
  const int ntile = blockIdx.y;
  const int ncol0 = ntile * 16;
  const int tid = threadIdx.x;

  // stage B[:, ncol0:ncol0+16] into LDS, zero-pad columns beyond N
  for (int t = tid; t < K * 16; t += blockDim.x) {
    int k = t >> 4, nn = t & 15;
    int col = ncol0 + nn;
    ldsB[t] = (col < N) ? B[(size_t)k * N + col] : 0.f;
  }
  __syncthreads();

  const int wave = tid >> 5;
  const int lane = tid & 31;
  const int half = lane >> 4;   // 0: K pair {0,1}, 1: K pair {2,3}
  const int l16 = lane & 15;
  const int mtile = blockIdx.x * GEMM_WAVES + wave;
  if (mtile * 16 >= M) return;

  const int row = mtile * 16 + l16;           // A row for this lane
  const float* arow = A + (size_t)row * K;
  v8f acc = {};
  const int nk = K >> 2;
  for (int kk = 0; kk < nk; ++kk) {
    const int k0 = kk * 4 + half * 2;
    float2 av = *(const float2*)(arow + k0);  // 8B-aligned (K mult of 4)
    v2f a, b;
    a.x = av.x;            // A[m][k0]
    a.y = av.y;            // A[m][k0+1]
    b.x = ldsB[k0 * 16 + l16];        // B[k0][n]
    b.y = ldsB[(k0 + 1) * 16 + l16];  // B[k0+1][n]
    acc = __builtin_amdgcn_wmma_f32_16x16x4_f32(
        false, a, false, b, (short)0, acc, false, false);
  }
  // D layout: VGPR r -> row (r + 8*half), col = lane&15
#pragma unroll
  for (int r = 0; r < 8; ++r) {
    int m = mtile * 16 + r + 8 * half;
    int col = ncol0 + l16;
    if (col < N) C[(size_t)m * N + col] = acc[r];
  }
}

// -------- per-node attention dots: s_src[n]=hw[n].a_src, s_dst likewise -----
__global__ void node_dots_kernel(const float* __restrict__ hw,
                                 const float* __restrict__ a_src,
                                 const float* __restrict__ a_dst,
                                 float* __restrict__ s_src,
                                 float* __restrict__ s_dst) {
  int wave = threadIdx.x >> 5, lane = threadIdx.x & 31;
  int n = blockIdx.x * (blockDim.x >> 5) + wave;
  if (n >= NNODES) return;
  const float* row = hw + (size_t)n * EMB;
  float ss = 0.f, sd = 0.f;
  for (int d = lane; d < EMB; d += 32) {
    float v = row[d];
    ss += v * a_src[d];
    sd += v * a_dst[d];
  }
  ss = warp_sum(ss);
  sd = warp_sum(sd);
  if (lane == 0) { s_src[n] = ss; s_dst[n] = sd; }
}

// -------- init node softmax state + bias-broadcast output buffer ------------
__global__ void node_init_kernel(unsigned* __restrict__ mkey,
                                 float* __restrict__ ssum) {
  int n = blockIdx.x * blockDim.x + threadIdx.x;
  if (n < NNODES) { mkey[n] = 0u; ssum[n] = 0.f; }
}
__global__ void bias_init_kernel(float* __restrict__ h,
                                 const float* __restrict__ b) {
  long i = (long)blockIdx.x * blockDim.x + threadIdx.x;
  if (i < (long)NNODES * EMB) h[i] = b[i % EMB];
}
__global__ void zero_kernel(float* __restrict__ p, long n) {
  long i = (long)blockIdx.x * blockDim.x + threadIdx.x;
  if (i < n) p[i] = 0.f;
}

// -------- pass 1: logit[e] = leaky(s_src[src]+s_dst[dst]+ee.a_edge); segmax --
__global__ void edge_logit_kernel(const int* __restrict__ src,
                                  const int* __restrict__ dst,
                                  const int* __restrict__ attr,
                                  const float* __restrict__ bond_emb,
                                  const float* __restrict__ a_edge,
                                  const float* __restrict__ s_src,
                                  const float* __restrict__ s_dst,
                                  float* __restrict__ logit,
                                  unsigned* __restrict__ mkey) {
  int wave = threadIdx.x >> 5, lane = threadIdx.x & 31;
  int e = blockIdx.x * (blockDim.x >> 5) + wave;
  if (e >= NEDGES) return;
  int i0 = attr[e * BONDF + 0], i1 = attr[e * BONDF + 1], i2 = attr[e * BONDF + 2];
  const float* t0 = bond_emb + (size_t)(0 * BONDV + i0) * EMB;
  const float* t1 = bond_emb + (size_t)(1 * BONDV + i1) * EMB;
  const float* t2 = bond_emb + (size_t)(2 * BONDV + i2) * EMB;
  float se = 0.f;
  for (int d = lane; d < EMB; d += 32) se += (t0[d] + t1[d] + t2[d]) * a_edge[d];
  se = warp_sum(se);
  if (lane == 0) {
    float lg = s_src[src[e]] + s_dst[dst[e]] + se;
    lg = lg > 0.f ? lg : NEG_SLOPE * lg;
    logit[e] = lg;
    atomicMax(&mkey[dst[e]], f2key(lg));
  }
}

// -------- pass 2: expv[e] = exp(logit - max[dst]); segsum into ssum ----------
__global__ void edge_exp_kernel(const int* __restrict__ dst,
                                const unsigned* __restrict__ mkey,
                                float* __restrict__ edgebuf,  // in: logit, out: exp
                                float* __restrict__ ssum) {
  int e = blockIdx.x * blockDim.x + threadIdx.x;
  if (e >= NEDGES) return;
  int t = dst[e];
  float ex = __expf(edgebuf[e] - key2f(mkey[t]));
  edgebuf[e] = ex;
  atomicAdd(&ssum[t], ex);
}

// -------- pass 3: hout[dst] += (hw[src] + ee) * alpha -----------------------
__global__ void edge_message_kernel(const int* __restrict__ src,
                                    const int* __restrict__ dst,
                                    const int* __restrict__ attr,
                                    const float* __restrict__ bond_emb,
                                    const float* __restrict__ hw,
                                    const float* __restrict__ expv,
                                    const float* __restrict__ ssum,
                                    float* __restrict__ hout) {
  int wave = threadIdx.x >> 5, lane = threadIdx.x & 31;
  int e = blockIdx.x * (blockDim.x >> 5) + wave;
  if (e >= NEDGES) return;
  int s = src[e], t = dst[e];
  float alpha = expv[e] / (ssum[t] + 1e-16f);
  int i0 = attr[e * BONDF + 0], i1 = attr[e * BONDF + 1], i2 = attr[e * BONDF + 2];
  const float* t0 = bond_emb + (size_t)(0 * BONDV + i0) * EMB;
  const float* t1 = bond_emb + (size_t)(1 * BONDV + i1) * EMB;
  const float* t2 = bond_emb + (size_t)(2 * BONDV + i2) * EMB;
  const float* hrow = hw + (size_t)s * EMB;
  float* orow = hout + (size_t)t * EMB;
  for (int d = lane; d < EMB; d += 32)
    atomicAdd(&orow[d], (hrow[d] + t0[d] + t1[d] + t2[d]) * alpha);
}

__global__ void relu_kernel(float* __restrict__ h, long n) {
  long i = (long)blockIdx.x * blockDim.x + threadIdx.x;
  if (i < n) h[i] = fmaxf(h[i], 0.f);
}

// -------- graph readout ------------------------------------------------------
__global__ void graph_sum_kernel(const float* __restrict__ h,
                                 const int* __restrict__ batch,
                                 float* __restrict__ g) {
  long i = (long)blockIdx.x * blockDim.x + threadIdx.x;
  if (i >= (long)NNODES * EMB) return;
  int n = (int)(i / EMB), d = (int)(i % EMB);
  atomicAdd(&g[(size_t)batch[n] * EMB + d], h[i]);
}

__global__ void final_kernel(const float* __restrict__ g,
                             const float* __restrict__ lin_w,
                             const float* __restrict__ lin_b,
                             float* __restrict__ out) {
  int wave = threadIdx.x >> 5, lane = threadIdx.x & 31;
  int j = blockIdx.x * (blockDim.x >> 5) + wave;
  if (j >= NGRAPHS) return;
  const float* row = g + (size_t)j * EMB;
  float s = 0.f;
  for (int d = lane; d < EMB; d += 32) s += row[d] * lin_w[d];
  s = warp_sum(s);
  if (lane == 0) out[j] = s + lin_b[0];
}

// ---------------------------------------------------------------------------
static inline int gdiv(long n, int b) { return (int)((n + b - 1) / b); }

extern "C" void kernel_launch(void* const* d_in, const int* in_sizes, int n_in,
                              void* d_out, int out_size, void* d_ws, size_t ws_size,
                              hipStream_t stream) {
  const int* x         = (const int*)d_in[0];
  const int* eidx      = (const int*)d_in[1];
  const int* eattr     = (const int*)d_in[2];
  const int* batch     = (const int*)d_in[3];
  const float* atom_emb  = (const float*)d_in[4];
  const float* bond_emb1 = (const float*)d_in[5];
  const float* W1        = (const float*)d_in[6];
  const float* a_src1    = (const float*)d_in[7];
  const float* a_dst1    = (const float*)d_in[8];
  const float* a_edge1   = (const float*)d_in[9];
  const float* b1        = (const float*)d_in[10];
  const float* bond_emb2 = (const float*)d_in[11];
  const float* W2        = (const float*)d_in[12];
  const float* a_src2    = (const float*)d_in[13];
  const float* a_dst2    = (const float*)d_in[14];
  const float* a_edge2   = (const float*)d_in[15];
  const float* b2        = (const float*)d_in[16];
  const float* lin_w     = (const float*)d_in[17];
  const float* lin_b     = (const float*)d_in[18];
  float* out = (float*)d_out;

  const int* src = eidx;
  const int* dst = eidx + NEDGES;

  // workspace carve-up (floats)
  float* ws = (float*)d_ws;
  size_t off = 0;
  float* bufA = ws + off; off += (size_t)NNODES * EMB;   // h / layer2 out
  float* bufB = ws + off; off += (size_t)NNODES * EMB;   // hw
  float* bufC = ws + off; off += (size_t)NNODES * EMB;   // layer1 out
  float* g    = ws + off; off += (size_t)NGRAPHS * EMB;
  float* s_src = ws + off; off += NNODES;
  float* s_dst = ws + off; off += NNODES;
  float* ssum  = ws + off; off += NNODES;
  unsigned* mkey = (unsigned*)(ws + off); off += NNODES;
  float* ebuf  = ws + off; off += NEDGES;                // logit -> exp

  const long NE = (long)NNODES * EMB;
  const dim3 gemm_grid(gdiv(NNODES / 16, GEMM_WAVES), gdiv(EMB, 16));

  // ---- layer 1 ----
  encode_atoms_kernel<<<NNODES, 128, 0, stream>>>(x, atom_emb, bufA);
  gemm_wmma_kernel<<<gemm_grid, 256, 0, stream>>>(bufA, W1, bufB, NNODES, EMB, EMB);
  node_dots_kernel<<<gdiv(NNODES, 8), 256, 0, stream>>>(bufB, a_src1, a_dst1, s_src, s_dst);
  node_init_kernel<<<gdiv(NNODES, 256), 256, 0, stream>>>(mkey, ssum);
  bias_init_kernel<<<gdiv(NE, 256), 256, 0, stream>>>(bufC, b1);
  edge_logit_kernel<<<gdiv(NEDGES, 8), 256, 0, stream>>>(src, dst, eattr, bond_emb1,
                                                         a_edge1, s_src, s_dst, ebuf, mkey);
  edge_exp_kernel<<<gdiv(NEDGES, 256), 256, 0, stream>>>(dst, mkey, ebuf, ssum);
  edge_message_kernel<<<gdiv(NEDGES, 8), 256, 0, stream>>>(src, dst, eattr, bond_emb1,
                                                           bufB, ebuf, ssum, bufC);
  relu_kernel<<<gdiv(NE, 256), 256, 0, stream>>>(bufC, NE);

  // ---- layer 2 ----
  gemm_wmma_kernel<<<gemm_grid, 256, 0, stream>>>(bufC, W2, bufB, NNODES, EMB, EMB);
  node_dots_kernel<<<gdiv(NNODES, 8), 256, 0, stream>>>(bufB, a_src2, a_dst2, s_src, s_dst);
  node_init_kernel<<<gdiv(NNODES, 256), 256, 0, stream>>>(mkey, ssum);
  bias_init_kernel<<<gdiv(NE, 256), 256, 0, stream>>>(bufA, b2);
  edge_logit_kernel<<<gdiv(NEDGES, 8), 256, 0, stream>>>(src, dst, eattr, bond_emb2,
                                                         a_edge2, s_src, s_dst, ebuf, mkey);
  edge_exp_kernel<<<gdiv(NEDGES, 256), 256, 0, stream>>>(dst, mkey, ebuf, ssum);
  edge_message_kernel<<<gdiv(NEDGES, 8), 256, 0, stream>>>(src, dst, eattr, bond_emb2,
                                                           bufB, ebuf, ssum, bufA);

  // ---- readout ----
  zero_kernel<<<gdiv((long)NGRAPHS * EMB, 256), 256, 0, stream>>>(g, (long)NGRAPHS * EMB);
  graph_sum_kernel<<<gdiv(NE, 256), 256, 0, stream>>>(bufA, batch, g);
  final_kernel<<<gdiv(NGRAPHS, 8), 256, 0, stream>>>(g, lin_w, lin_b, out);
}